// SCGSnn_6846177870000
// MI455X (gfx1250) — compile-verified
//
#include <hip/hip_runtime.h>

typedef float v2f __attribute__((ext_vector_type(2)));
typedef float v8f __attribute__((ext_vector_type(8)));

#define BETA 0.9f
#define THR  1.0f
#define TSTEPS 32
#define N_INF 256          // n_in
#define N_H   64           // hidden
#define N_C   3            // classes
#define ROWS_PER_BLOCK 128 // 4 waves * 32 rows
#define LDS_STRIDE 65      // 64 + 1 pad -> conflict-free column reads

__global__ __launch_bounds__(ROWS_PER_BLOCK)
void snn_fused_wmma_kernel(const float* __restrict__ x,
                           const float* __restrict__ W1,
                           const float* __restrict__ W2,
                           float* __restrict__ out)
{
    __shared__ float sI1[ROWS_PER_BLOCK * LDS_STRIDE];
    __shared__ float sW2[N_C * N_H];

    const int tid  = threadIdx.x;
    const int w    = tid >> 5;      // wave id (wave32)
    const int lane = tid & 31;
    const int half = lane >> 4;     // half-wave select (K split)
    const int l16  = lane & 15;     // M / N index within tile
    const long long blockRow = (long long)blockIdx.x * ROWS_PER_BLOCK;

    // ---------------- Phase 1: I1 = x @ W1^T  (f32 WMMA 16x16x4) ----------------
    // Wave handles 32 rows: 2 M-tiles x 4 N-tiles, K swept 4 at a time.
    v8f acc[2][4];
    #pragma unroll
    for (int rt = 0; rt < 2; ++rt)
        #pragma unroll
        for (int nt = 0; nt < 4; ++nt)
            #pragma unroll
            for (int j = 0; j < 8; ++j)
                acc[rt][nt][j] = 0.0f;

    const long long rowA0 = blockRow + (long long)w * 32 + l16;       // M-tile 0
    const long long rowA1 = rowA0 + 16;                                // M-tile 1

    for (int k = 0; k < N_INF; k += 4) {
        // A: 16x4 f32 tile: lanes 0-15 hold K=k..k+1, lanes 16-31 hold K=k+2..k+3
        v2f a0 = *(const v2f*)(x + rowA0 * N_INF + k + 2 * half);
        v2f a1 = *(const v2f*)(x + rowA1 * N_INF + k + 2 * half);
        // B: 4x16 tile of W1^T -> contiguous K pairs from W1 rows (same pattern)
        v2f b[4];
        #pragma unroll
        for (int nt = 0; nt < 4; ++nt)
            b[nt] = *(const v2f*)(W1 + (long long)(nt * 16 + l16) * N_INF + k + 2 * half);

        #pragma unroll
        for (int nt = 0; nt < 4; ++nt) {
            acc[0][nt] = __builtin_amdgcn_wmma_f32_16x16x4_f32(
                false, a0, false, b[nt], (short)0, acc[0][nt], false, false);
            acc[1][nt] = __builtin_amdgcn_wmma_f32_16x16x4_f32(
                false, a1, false, b[nt], (short)0, acc[1][nt], false, false);
        }
    }

    // Stage I1 tile to LDS (C layout: VGPR j -> M=j (lanes 0-15), M=j+8 (lanes 16-31))
    #pragma unroll
    for (int rt = 0; rt < 2; ++rt)
        #pragma unroll
        for (int nt = 0; nt < 4; ++nt)
            #pragma unroll
            for (int j = 0; j < 8; ++j) {
                int rl  = w * 32 + rt * 16 + j + 8 * half;  // row within block
                int col = nt * 16 + l16;                    // hidden index
                sI1[rl * LDS_STRIDE + col] = acc[rt][nt][j];
            }

    for (int i = tid; i < N_C * N_H; i += ROWS_PER_BLOCK)
        sW2[i] = W2[i];
    __syncthreads();

    // ---------------- Phase 2: LIF recurrence, thread = one batch row ----------------
    // I1 is time-constant, so loop h outer / t inner and accumulate the
    // per-timestep layer-2 currents J[t][c] in registers (96 VGPRs).
    float J[TSTEPS][N_C];
    #pragma unroll
    for (int t = 0; t < TSTEPS; ++t) {
        J[t][0] = 0.0f; J[t][1] = 0.0f; J[t][2] = 0.0f;
    }

    const float* myI1 = sI1 + tid * LDS_STRIDE;  // stride 65 -> conflict-free
    for (int h = 0; h < N_H; ++h) {
        float cur = myI1[h];
        float w2a = sW2[h];
        float w2b = sW2[N_H + h];
        float w2c = sW2[2 * N_H + h];
        float v1 = 0.0f;
        #pragma unroll
        for (int t = 0; t < TSTEPS; ++t) {
            v1 = __builtin_fmaf(BETA, v1, cur);          // v1 = beta*v1 + I1
            float s = (v1 >= THR) ? 1.0f : 0.0f;         // Heaviside(v1 - THR)
            v1 -= s;                                     // reset by THR (=1)
            J[t][0] = __builtin_fmaf(s, w2a, J[t][0]);   // s1 @ W2^T, class 0
            J[t][1] = __builtin_fmaf(s, w2b, J[t][1]);
            J[t][2] = __builtin_fmaf(s, w2c, J[t][2]);
        }
    }

    // Layer-2 recurrence + spike count (pure registers)
    float v2[N_C] = {0.0f, 0.0f, 0.0f};
    float sc[N_C] = {0.0f, 0.0f, 0.0f};
    #pragma unroll
    for (int t = 0; t < TSTEPS; ++t) {
        #pragma unroll
        for (int c = 0; c < N_C; ++c) {
            v2[c] = __builtin_fmaf(BETA, v2[c], J[t][c]);
            float s2 = (v2[c] >= THR) ? 1.0f : 0.0f;
            v2[c] -= s2;
            sc[c] += s2;
        }
    }

    const long long row = blockRow + tid;
    out[row * 3 + 0] = sc[0];
    out[row * 3 + 1] = sc[1];
    out[row * 3 + 2] = sc[2];
}

extern "C" void kernel_launch(void* const* d_in, const int* in_sizes, int n_in,
                              void* d_out, int out_size, void* d_ws, size_t ws_size,
                              hipStream_t stream) {
    (void)n_in; (void)out_size; (void)d_ws; (void)ws_size;
    const float* x  = (const float*)d_in[0];   // (B, 256)
    const float* W1 = (const float*)d_in[1];   // (64, 256)
    const float* W2 = (const float*)d_in[2];   // (3, 64)
    float* out = (float*)d_out;                // (B, 3)

    const int B = in_sizes[0] / N_INF;         // 131072
    const int blocks = B / ROWS_PER_BLOCK;     // 1024
    snn_fused_wmma_kernel<<<blocks, ROWS_PER_BLOCK, 0, stream>>>(x, W1, W2, out);
}